// mHCGenerator_80951543595212
// MI455X (gfx1250) — compile-verified
//
#include <hip/hip_runtime.h>
#include <math.h>

typedef __attribute__((ext_vector_type(2))) float v2f;
typedef __attribute__((ext_vector_type(8))) float v8f;

#define D_FLAT     4096
#define SK_ITERS   20
#define SK_EPS     1e-6f
#define RMS_EPS    1.1920928955078125e-07f   // np.finfo(float32).eps

#define ROWS_PER_WG 128
#define THREADS     256
#define KSLAB       64
#define XS_STRIDE   68   // padded row stride (words) -> conflict-free ds_load_b64
#define WS_STRIDE   68

__global__ __launch_bounds__(THREADS)
void mhc_fused_kernel(const float* __restrict__ x,
                      const float* __restrict__ norm_w,
                      const float* __restrict__ Wpre,
                      const float* __restrict__ Wpost,
                      const float* __restrict__ Wres,
                      const float* __restrict__ bpre,
                      const float* __restrict__ bpost,
                      const float* __restrict__ bres,
                      const float* __restrict__ apre_p,
                      const float* __restrict__ apost_p,
                      const float* __restrict__ ares_p,
                      float* __restrict__ out, int Btot)
{
    __shared__ float xs[ROWS_PER_WG * XS_STRIDE];   // 34816 B
    __shared__ float wsl[32 * WS_STRIDE];           //  8704 B
    __shared__ float resl[ROWS_PER_WG * 24];        // 12288 B

    const int t    = threadIdx.x;
    const int lane = t & 31;
    const int wave = t >> 5;
    const int half = lane >> 4;   // 0 or 1 (half-wave)
    const int l16  = lane & 15;
    const int row0 = blockIdx.x * ROWS_PER_WG;

    v8f acc0 = {};   // output cols 0..15  (pre 0..3, post 4..7, res 8..15)
    v8f acc1 = {};   // output cols 16..31 (res 16..23, 24..31 = zero pad)
    float ss = 0.0f; // partial sum of squares for RMSNorm

    // A fragment (x): lane L<16 -> row L, K = kk+{0,1}; lane L+16 -> row L, K = kk+{2,3}
    const int aoff  = (wave * 16 + l16) * XS_STRIDE + 2 * half;
    // B fragment (weights): lane n<16 -> col n, K = kk+{0,1}; lane n+16 -> col n, K = kk+{2,3}
    const int b0off = l16 * WS_STRIDE + 2 * half;
    const int b1off = (16 + l16) * WS_STRIDE + 2 * half;

    // x-slab loader indices: 16 lanes cover one row's 64 floats (fully coalesced b128)
    const int xrl_base = wave * 2 + half;   // 0..15
    const int xkl      = l16 * 4;
    // weight-slab loader indices
    const int wk = t & 63;
    const int jb = t >> 6;                  // 0..3

    for (int k0 = 0; k0 < D_FLAT; k0 += KSLAB) {
        // ---- stage x slab: 128 rows x 64 k ----
        #pragma unroll
        for (int i = 0; i < 8; ++i) {
            const int rl = i * 16 + xrl_base;
            const float4 v = *(const float4*)(x + (size_t)(row0 + rl) * D_FLAT + k0 + xkl);
            *(float4*)(&xs[rl * XS_STRIDE + xkl]) = v;
        }
        // ---- stage composed weight slab: Wt[j][k] = norm_w[k] * W_j[k], j=24..31 zero ----
        {
            const float nw = norm_w[k0 + wk];
            #pragma unroll
            for (int jt = 0; jt < 8; ++jt) {
                const int j = jt * 4 + jb;   // uniform per wave -> no divergence
                float w = 0.0f;
                if (j < 4)       w = Wpre [(size_t)j       * D_FLAT + k0 + wk];
                else if (j < 8)  w = Wpost[(size_t)(j - 4) * D_FLAT + k0 + wk];
                else if (j < 24) w = Wres [(size_t)(j - 8) * D_FLAT + k0 + wk];
                wsl[j * WS_STRIDE + wk] = w * nw;
            }
        }
        __syncthreads();

        // ---- WMMA K loop over the slab ----
        #pragma unroll
        for (int kk = 0; kk < KSLAB; kk += 4) {
            const v2f a  = *(const v2f*)(&xs [aoff  + kk]);
            const v2f b0 = *(const v2f*)(&wsl[b0off + kk]);
            const v2f b1 = *(const v2f*)(&wsl[b1off + kk]);
            acc0 = __builtin_amdgcn_wmma_f32_16x16x4_f32(false, a, false, b0,
                                                         (short)0, acc0, false, false);
            acc1 = __builtin_amdgcn_wmma_f32_16x16x4_f32(false, a, false, b1,
                                                         (short)0, acc1, false, false);
            ss = fmaf(a.x, a.x, ss);
            ss = fmaf(a.y, a.y, ss);
        }
        __syncthreads();
    }

    // ---- RMS scale: combine half-wave partials, rsqrt, apply to accumulators ----
    ss += __shfl_xor(ss, 16, 32);                       // lane L & L+16 -> full row sum
    const float rn = rsqrtf(ss * (1.0f / (float)D_FLAT) + RMS_EPS);

    #pragma unroll
    for (int i = 0; i < 8; ++i) {
        const float rni = __shfl(rn, i + 8 * half, 32); // rn of row M = i + 8*half
        const int rl = wave * 16 + i + 8 * half;
        resl[rl * 24 + l16] = acc0[i] * rni;            // cols 0..15
        if (l16 < 8)
            resl[rl * 24 + 16 + l16] = acc1[i] * rni;   // cols 16..23
    }
    __syncthreads();

    // ---- per-row epilogue: sigmoids + 4x4 Sinkhorn-Knopp ----
    if (t < ROWS_PER_WG) {
        const int rg = row0 + t;
        const float a_pre  = apre_p[0];
        const float a_post = apost_p[0];
        const float a_res  = ares_p[0];

        float v[24];
        #pragma unroll
        for (int j = 0; j < 24; ++j) v[j] = resl[t * 24 + j];

        // H_pre = sigmoid(alpha_pre * dot + b_pre)
        {
            float4 r;
            r.x = 1.0f / (1.0f + expf(-(a_pre * v[0] + bpre[0])));
            r.y = 1.0f / (1.0f + expf(-(a_pre * v[1] + bpre[1])));
            r.z = 1.0f / (1.0f + expf(-(a_pre * v[2] + bpre[2])));
            r.w = 1.0f / (1.0f + expf(-(a_pre * v[3] + bpre[3])));
            *(float4*)(out + (size_t)rg * 4) = r;
        }
        // H_post = 2 * sigmoid(alpha_post * dot + b_post)
        {
            float4 r;
            r.x = 2.0f / (1.0f + expf(-(a_post * v[4] + bpost[0])));
            r.y = 2.0f / (1.0f + expf(-(a_post * v[5] + bpost[1])));
            r.z = 2.0f / (1.0f + expf(-(a_post * v[6] + bpost[2])));
            r.w = 2.0f / (1.0f + expf(-(a_post * v[7] + bpost[3])));
            *(float4*)(out + (size_t)Btot * 4 + (size_t)rg * 4) = r;
        }
        // H_res = sinkhorn_knopp(exp(alpha_res * dot + b_res)) on 4x4
        float m[16];
        #pragma unroll
        for (int j = 0; j < 16; ++j) m[j] = expf(a_res * v[8 + j] + bres[j]);

        for (int it = 0; it < SK_ITERS; ++it) {
            #pragma unroll
            for (int r = 0; r < 4; ++r) {               // row normalize (axis=-1)
                const float s = m[r*4] + m[r*4+1] + m[r*4+2] + m[r*4+3] + SK_EPS;
                const float inv = 1.0f / s;
                m[r*4]   *= inv; m[r*4+1] *= inv;
                m[r*4+2] *= inv; m[r*4+3] *= inv;
            }
            #pragma unroll
            for (int c = 0; c < 4; ++c) {               // col normalize (axis=-2)
                const float s = m[c] + m[4+c] + m[8+c] + m[12+c] + SK_EPS;
                const float inv = 1.0f / s;
                m[c] *= inv; m[4+c] *= inv; m[8+c] *= inv; m[12+c] *= inv;
            }
        }
        float* outr = out + (size_t)Btot * 8 + (size_t)rg * 16;
        #pragma unroll
        for (int j = 0; j < 16; j += 4) {
            float4 r; r.x = m[j]; r.y = m[j+1]; r.z = m[j+2]; r.w = m[j+3];
            *(float4*)(outr + j) = r;
        }
    }
}

extern "C" void kernel_launch(void* const* d_in, const int* in_sizes, int n_in,
                              void* d_out, int out_size, void* d_ws, size_t ws_size,
                              hipStream_t stream)
{
    const float* x      = (const float*)d_in[0];
    const float* norm_w = (const float*)d_in[1];
    const float* Wpre   = (const float*)d_in[2];
    const float* Wpost  = (const float*)d_in[3];
    const float* Wres   = (const float*)d_in[4];
    const float* bpre   = (const float*)d_in[5];
    const float* bpost  = (const float*)d_in[6];
    const float* bres   = (const float*)d_in[7];
    const float* apre   = (const float*)d_in[8];
    const float* apost  = (const float*)d_in[9];
    const float* ares   = (const float*)d_in[10];

    const int Btot = in_sizes[0] / D_FLAT;        // 32768
    const int grid = Btot / ROWS_PER_WG;          // 256 workgroups

    mhc_fused_kernel<<<grid, THREADS, 0, stream>>>(
        x, norm_w, Wpre, Wpost, Wres, bpre, bpost, bres,
        apre, apost, ares, (float*)d_out, Btot);
}